// RoleContrastLossOriginalEPAL_79671643341678
// MI455X (gfx1250) — compile-verified
//
#include <hip/hip_runtime.h>
#include <hip/hip_bf16.h>

typedef float v2f __attribute__((ext_vector_type(2)));
typedef float v8f __attribute__((ext_vector_type(8)));

namespace {
constexpr int NB  = 64;    // batch
constexpr int NER = 1024;  // E*R slices
constexpr int NH  = 256;   // hidden
constexpr int KC  = 128;   // K chunk staged in LDS
constexpr int LDP = 132;   // padded LDS row stride (dwords): 132 % 64 == 4 -> conflict-free
constexpr int SLICES_PER_BLOCK = 4;   // 256 blocks x 4 slices: fill the big WGP array
constexpr int NBLOCKS = NER / SLICES_PER_BLOCK;
}

// ---------------- norms: one wave32 per (b,e,r) row ----------------
__global__ __launch_bounds__(256) void norms_kernel(const float* __restrict__ x,
                                                    float* __restrict__ norms) {
    int gid  = blockIdx.x * blockDim.x + threadIdx.x;
    int row  = gid >> 5;          // 0 .. 65535
    int lane = gid & 31;
    const float4* p = (const float4*)(x + (size_t)row * NH);
    float4 a = p[lane * 2];
    float4 b = p[lane * 2 + 1];
    float s = a.x*a.x + a.y*a.y + a.z*a.z + a.w*a.w
            + b.x*b.x + b.y*b.y + b.z*b.z + b.w*b.w;
    for (int m = 16; m >= 1; m >>= 1) s += __shfl_xor(s, m, 32);
    if (lane == 0) norms[row] = sqrtf(s);
}

// ---------------- zero the accumulators (ws is poisoned) ----------------
__global__ void zero_kernel(float* __restrict__ f, float* __restrict__ g,
                            int* __restrict__ sv) {
    int i = blockIdx.x * blockDim.x + threadIdx.x;
    if (i < NB * NB) { f[i] = 0.0f; g[i] = 0.0f; sv[i] = 0; }
}

// ---------------- main: Gram via V_WMMA_F32_16X16X4_F32 + fused epilogue ----------------
__global__ __launch_bounds__(256) void gram_kernel(const float* __restrict__ x,
                                                   const float* __restrict__ norms,
                                                   const int*   __restrict__ label,
                                                   float* __restrict__ fsum,
                                                   float* __restrict__ gsum,
                                                   int*   __restrict__ sany_g) {
    __shared__ float lds[NB * LDP];
    __shared__ float snorm[NB];
    __shared__ int   slabel[NB];

    const int tid  = threadIdx.x;
    const int wave = tid >> 5;
    const int lane = tid & 31;
    const int hs   = lane >> 4;   // half-select (lanes 0-15 vs 16-31)
    const int m    = lane & 15;

    const int ti  = wave >> 1;            // tile row 0..3 (i block)
    const int tj0 = (wave & 1) * 2;       // tile cols tj0, tj0+1 (j blocks)

    float facc0[8] = {}, facc1[8] = {}, gacc0[8] = {}, gacc1[8] = {};
    int sbits0 = 0, sbits1 = 0;

    for (int s = 0; s < SLICES_PER_BLOCK; ++s) {
        const int er = blockIdx.x * SLICES_PER_BLOCK + s;
        const float* xbase = x + (size_t)er * NH;  // row i at xbase + i*NER*NH

        v8f acc0 = {0.f,0.f,0.f,0.f,0.f,0.f,0.f,0.f};
        v8f acc1 = {0.f,0.f,0.f,0.f,0.f,0.f,0.f,0.f};

        for (int chunk = 0; chunk < 2; ++chunk) {
            const int k0 = chunk * KC;
            __syncthreads();   // previous LDS consumers done
            // stage 64 x 128 fp32 chunk: 2048 float4, 8 per thread
            #pragma unroll
            for (int it = 0; it < 8; ++it) {
                int q   = tid + it * 256;
                int row = q >> 5;
                int c4  = (q & 31) * 4;
                const float4 v = *(const float4*)(xbase + (size_t)row * (NER * NH) + k0 + c4);
                float* d = &lds[row * LDP + c4];
                d[0] = v.x; d[1] = v.y; d[2] = v.z; d[3] = v.w;
            }
            if (chunk == 0 && tid < NB) {
                snorm[tid]  = norms[(size_t)tid * NER + er];
                slabel[tid] = label[(size_t)tid * NER + er];
            }
            __syncthreads();

            const float* ra  = &lds[(ti * 16 + m)        * LDP + 2 * hs];
            const float* rb0 = &lds[(tj0 * 16 + m)       * LDP + 2 * hs];
            const float* rb1 = &lds[((tj0 + 1) * 16 + m) * LDP + 2 * hs];
            for (int k = 0; k < KC; k += 4) {
                v2f a, b0, b1;
                a.x  = ra[k];  a.y  = ra[k + 1];
                b0.x = rb0[k]; b0.y = rb0[k + 1];
                b1.x = rb1[k]; b1.y = rb1[k + 1];
                acc0 = __builtin_amdgcn_wmma_f32_16x16x4_f32(
                           false, a, false, b0, (short)0, acc0, false, false);
                acc1 = __builtin_amdgcn_wmma_f32_16x16x4_f32(
                           false, a, false, b1, (short)0, acc1, false, false);
            }
        }

        // fused epilogue for this slice: sim -> exp -> masked accumulation
        const int jj0 = tj0 * 16 + m;
        const int jj1 = jj0 + 16;
        const float nj0 = snorm[jj0], nj1 = snorm[jj1];
        const int   lj0 = slabel[jj0], lj1 = slabel[jj1];
        #pragma unroll
        for (int c = 0; c < 8; ++c) {
            const int   i  = ti * 16 + c + 8 * hs;
            const float ni = snorm[i];
            const int   li = slabel[i];
            {
                float sim = acc0[c] / fmaxf(ni * nj0, 1e-8f);
                float ex  = expf(sim * 10.0f);
                bool same  = (li == lj0) && (li != 0) && (i != jj0);
                bool valid = (ni > 1e-6f) && (nj0 > 1e-6f);
                if (same)       { facc0[c] += ex; sbits0 |= (1 << c); }
                else if (valid) { gacc0[c] += ex; }
            }
            {
                float sim = acc1[c] / fmaxf(ni * nj1, 1e-8f);
                float ex  = expf(sim * 10.0f);
                bool same  = (li == lj1) && (li != 0) && (i != jj1);
                bool valid = (ni > 1e-6f) && (nj1 > 1e-6f);
                if (same)       { facc1[c] += ex; sbits1 |= (1 << c); }
                else if (valid) { gacc1[c] += ex; }
            }
        }
    }

    // one atomic round per block for lane-owned (i,j) positions
    const int jj0 = tj0 * 16 + m, jj1 = jj0 + 16;
    #pragma unroll
    for (int c = 0; c < 8; ++c) {
        const int i = ti * 16 + c + 8 * hs;
        atomicAdd(&fsum[i * NB + jj0], facc0[c]);
        atomicAdd(&gsum[i * NB + jj0], gacc0[c]);
        if (sbits0 & (1 << c)) atomicOr(&sany_g[i * NB + jj0], 1);
        atomicAdd(&fsum[i * NB + jj1], facc1[c]);
        atomicAdd(&gsum[i * NB + jj1], gacc1[c]);
        if (sbits1 & (1 << c)) atomicOr(&sany_g[i * NB + jj1], 1);
    }
}

// ---------------- final scalar reduction over 64x64 pairs ----------------
__global__ __launch_bounds__(256) void finalize_kernel(const float* __restrict__ fsum,
                                                       const float* __restrict__ gsum,
                                                       const int*   __restrict__ sany,
                                                       float* __restrict__ out) {
    __shared__ float stot[256];
    __shared__ int   scnt[256];
    const int t = threadIdx.x;
    float tot = 0.0f; int cnt = 0;
    for (int p = t; p < NB * NB; p += 256) {
        if (sany[p]) {
            float fv = fsum[p], gv = gsum[p];
            tot += -logf(fv / fmaxf(fv + gv, 1e-8f));
            cnt++;
        }
    }
    stot[t] = tot; scnt[t] = cnt;
    __syncthreads();
    for (int off = 128; off >= 1; off >>= 1) {
        if (t < off) { stot[t] += stot[t + off]; scnt[t] += scnt[t + off]; }
        __syncthreads();
    }
    if (t == 0) out[0] = (scnt[0] > 0) ? stot[0] / (float)scnt[0] : 0.0f;
}

extern "C" void kernel_launch(void* const* d_in, const int* in_sizes, int n_in,
                              void* d_out, int out_size, void* d_ws, size_t ws_size,
                              hipStream_t stream) {
    const float* x     = (const float*)d_in[0];   // (64,64,16,256) fp32
    const int*   label = (const int*)d_in[1];     // (64,64,16) int

    float* norms = (float*)d_ws;                  // 65536 floats
    float* fsum  = norms + NB * NER;              // 4096 floats
    float* gsum  = fsum + NB * NB;                // 4096 floats
    int*   sany  = (int*)(gsum + NB * NB);        // 4096 ints

    zero_kernel<<<16, 256, 0, stream>>>(fsum, gsum, sany);
    norms_kernel<<<(NB * NER) / 8, 256, 0, stream>>>(x, norms);
    gram_kernel<<<NBLOCKS, 256, 0, stream>>>(x, norms, label, fsum, gsum, sany);
    finalize_kernel<<<1, 256, 0, stream>>>(fsum, gsum, sany, (float*)d_out);
}